// DISL_Loss_63969242906738
// MI455X (gfx1250) — compile-verified
//
#include <hip/hip_runtime.h>
#include <hip/hip_bf16.h>
#include <math.h>

#define B_  64
#define T_  256
#define M_  1024
#define OM_ 512
#define N_  (B_ * T_)   // 16384

typedef float v2f __attribute__((ext_vector_type(2)));
typedef float v8f __attribute__((ext_vector_type(8)));

// ---------------------------------------------------------------- zero init
__global__ void zero_kernel(float* __restrict__ p, int n) {
  for (int i = blockIdx.x * blockDim.x + threadIdx.x; i < n; i += gridDim.x * blockDim.x)
    p[i] = 0.f;
}

// ------------------------------------------------- column sum-of-squares
// X: [N_, ncols] row-major. grid = (ncols/256, 16). atomicAdd partials.
__global__ __launch_bounds__(256) void colsumsq_kernel(const float* __restrict__ X,
                                                       float* __restrict__ ssq, int ncols) {
  const int j = blockIdx.x * 256 + threadIdx.x;
  const int rows = N_ / 16;
  const float* p = X + (size_t)(blockIdx.y * rows) * ncols + j;
  float s = 0.f;
  for (int k = 0; k < rows; ++k) {
    float x = p[(size_t)k * ncols];
    s += x * x;
  }
  atomicAdd(&ssq[j], s);
}

// --------------------------------------------------------- WMMA f32 GEMM
// sim[i,j] = (sum_k O[k,i]*V[k,j]) / (max(||O_i||,1e-12)*max(||V_j||,1e-12))
// O: [N_, om], V: [N_, M_], sim: [om, M_].
// One wave computes a 32x32 output block as a 2x2 grid of 16x16 WMMA tiles:
// 8 dword loads -> 4 WMMAs per K-step, halves L2 panel re-reads.
__global__ __launch_bounds__(256) void gemm_wmma_kernel(const float* __restrict__ O,
                                                        const float* __restrict__ V,
                                                        const float* __restrict__ ssO,
                                                        const float* __restrict__ ssV,
                                                        float* __restrict__ sim, int om) {
  const int wave = blockIdx.x * (blockDim.x / 32) + (threadIdx.x / 32);
  const int lane = threadIdx.x & 31;
  const int tilesJ = M_ / 32;     // 32
  const int ti = wave / tilesJ;   // wave-uniform -> EXEC stays all-ones
  const int tj = wave % tilesJ;
  if (ti >= om / 32) return;
  const int i0 = ti * 32, j0 = tj * 32;
  const int r  = lane & 15;          // row (A) / col (B) within 16x16 tile
  const int kb = (lane >> 4) << 1;   // 0 or 2 : K sub-slot

  const float* Oc = O + i0 + r;      // A[r][kb+q] = O[(k0+kb+q)*om + i0+r]
  const float* Vc = V + j0 + r;      // B[kb+q][n] = V[(k0+kb+q)*M_ + j0+n]
  v8f acc00 = {}, acc01 = {}, acc10 = {}, acc11 = {};
#pragma unroll 2
  for (int k0 = 0; k0 < N_; k0 += 4) {
    const size_t ko0 = (size_t)(k0 + kb) * om;
    const size_t ko1 = (size_t)(k0 + kb + 1) * om;
    const size_t kv0 = (size_t)(k0 + kb) * M_;
    const size_t kv1 = (size_t)(k0 + kb + 1) * M_;
    v2f a0, a1, b0, b1;
    a0.x = Oc[ko0];      a0.y = Oc[ko1];
    a1.x = Oc[ko0 + 16]; a1.y = Oc[ko1 + 16];
    b0.x = Vc[kv0];      b0.y = Vc[kv1];
    b1.x = Vc[kv0 + 16]; b1.y = Vc[kv1 + 16];
    acc00 = __builtin_amdgcn_wmma_f32_16x16x4_f32(false, a0, false, b0,
                                                  (short)0, acc00, false, false);
    acc01 = __builtin_amdgcn_wmma_f32_16x16x4_f32(false, a0, false, b1,
                                                  (short)0, acc01, false, false);
    acc10 = __builtin_amdgcn_wmma_f32_16x16x4_f32(false, a1, false, b0,
                                                  (short)0, acc10, false, false);
    acc11 = __builtin_amdgcn_wmma_f32_16x16x4_f32(false, a1, false, b1,
                                                  (short)0, acc11, false, false);
  }
  // epilogue: fold in the l2-normalization scales.
  // D layout (16x16 f32): VGPR v holds row (v + 8*(lane/16)), col lane%16.
  const int c0 = j0 + r;
  const int c1 = c0 + 16;
  const float nv0 = fmaxf(sqrtf(ssV[c0]), 1e-12f);
  const float nv1 = fmaxf(sqrtf(ssV[c1]), 1e-12f);
  const int rb0 = i0 + ((lane >> 4) << 3);
  const int rb1 = rb0 + 16;
#pragma unroll
  for (int v = 0; v < 8; ++v) {
    const int row0 = rb0 + v;
    const int row1 = rb1 + v;
    const float no0 = fmaxf(sqrtf(ssO[row0]), 1e-12f);
    const float no1 = fmaxf(sqrtf(ssO[row1]), 1e-12f);
    sim[(size_t)row0 * M_ + c0] = acc00[v] / (no0 * nv0);
    sim[(size_t)row0 * M_ + c1] = acc01[v] / (no0 * nv1);
    sim[(size_t)row1 * M_ + c0] = acc10[v] / (no1 * nv0);
    sim[(size_t)row1 * M_ + c1] = acc11[v] / (no1 * nv1);
  }
}

// -------------------------------------------- greedy matching + ext build
// Per block (blockIdx selects simA/simF): 512 sequential argmax rounds over
// 1024 columns, masking used columns (== argsort + first-unused, same
// smallest-index tie-break). Then tail-fill ext[512..1023] with unused cols.
__global__ __launch_bounds__(1024) void match_kernel(const float* __restrict__ s0,
                                                     const float* __restrict__ s1,
                                                     int* __restrict__ e0,
                                                     int* __restrict__ e1) {
  const float* sim = blockIdx.x ? s1 : s0;
  int* ext = blockIdx.x ? e1 : e0;
  __shared__ float sval[1024];
  __shared__ int   sidx[1024];
  __shared__ unsigned char used[1024];
  const int tid = threadIdx.x;
  used[tid] = 0;
  __syncthreads();
  for (int i = 0; i < OM_; ++i) {
    sval[tid] = used[tid] ? -__builtin_inff() : sim[(size_t)i * M_ + tid];
    sidx[tid] = tid;
    __syncthreads();
    for (int s = 512; s > 0; s >>= 1) {
      if (tid < s) {
        float ov = sval[tid + s]; int oi = sidx[tid + s];
        if (ov > sval[tid] || (ov == sval[tid] && oi < sidx[tid])) {
          sval[tid] = ov; sidx[tid] = oi;
        }
      }
      __syncthreads();
    }
    const int best = sidx[0];
    __syncthreads();                   // everyone read best before next round's writes
    if (tid == best) used[tid] = 1;
    if (tid == 0)    ext[i] = best;
  }
  __syncthreads();
  if (tid == 0) {
    int cnt = 0;
    for (int c = 0; c < M_; ++c)
      if (!used[c]) ext[OM_ + cnt++] = c;
  }
}

// ----------------------------------------------------------- cosine losses
// One block per (b,t) row. pA[j] = (extA[j]<512) ? OA[row, extA[j]] : 0.
__global__ __launch_bounds__(256) void cos_loss_kernel(const float* __restrict__ V,
                                                       const float* __restrict__ OA,
                                                       const float* __restrict__ OF,
                                                       const int* __restrict__ extA,
                                                       const int* __restrict__ extF,
                                                       float* __restrict__ bank) {
  const int row = blockIdx.x;
  const float* vr = V  + (size_t)row * M_;
  const float* ar = OA + (size_t)row * OM_;
  const float* fr = OF + (size_t)row * OM_;
  float vv = 0, aa = 0, ff = 0, va = 0, vf = 0, af = 0;
  for (int j = threadIdx.x; j < M_; j += 256) {
    const float v = vr[j];
    const int ea = extA[j]; const float a = (ea < OM_) ? ar[ea] : 0.f;
    const int ef = extF[j]; const float f = (ef < OM_) ? fr[ef] : 0.f;
    vv += v * v; aa += a * a; ff += f * f;
    va += v * a; vf += v * f; af += a * f;
  }
  __shared__ float red[6][256];
  const int tid = threadIdx.x;
  red[0][tid] = vv; red[1][tid] = aa; red[2][tid] = ff;
  red[3][tid] = va; red[4][tid] = vf; red[5][tid] = af;
  __syncthreads();
  for (int s = 128; s > 0; s >>= 1) {
    if (tid < s)
      for (int q = 0; q < 6; ++q) red[q][tid] += red[q][tid + s];
    __syncthreads();
  }
  if (tid == 0) {
    const float nv = sqrtf(red[0][0]), na = sqrtf(red[1][0]), nf = sqrtf(red[2][0]);
    const float cA  = red[3][0] / fmaxf(nv * na, 1e-8f);
    const float cF  = red[4][0] / fmaxf(nv * nf, 1e-8f);
    const float cAF = red[5][0] / fmaxf(na * nf, 1e-8f);
    const int bk = blockIdx.x & 63;
    atomicAdd(&bank[0 * 64 + bk], 1.f - cA);
    atomicAdd(&bank[1 * 64 + bk], 1.f - cF);
    atomicAdd(&bank[2 * 64 + bk], 1.f - cAF);
  }
}

// ----------------------------------------------------------------- CE terms
__device__ __forceinline__ float ce_term(float q, float p) {
  const float eps = 1e-6f;
  p = fminf(fmaxf(p, eps), 1.f - eps);
  q = fminf(fmaxf(q, eps), 1.f - eps);
  return -(p * logf(q) + (1.f - p) * logf(1.f - q));
}

__global__ __launch_bounds__(256) void ce_kernel(const float* __restrict__ v_avf,
                                                 const float* __restrict__ va_avf,
                                                 const float* __restrict__ vf_avf,
                                                 const int* __restrict__ seq_len,
                                                 float* __restrict__ bank) {
  const int e = blockIdx.x * 256 + threadIdx.x;   // 64 blocks -> 16384 elems
  const int b = e >> 8, t = e & 255;
  const float m = (t < seq_len[b]) ? 1.f : 0.f;
  const float Vs = v_avf[e] * m, As = va_avf[e] * m, Fs = vf_avf[e] * m;
  float c0 = ce_term(Vs, As), c1 = ce_term(Vs, Fs), c2 = ce_term(As, Fs);
  __shared__ float red[3][256];
  const int tid = threadIdx.x;
  red[0][tid] = c0; red[1][tid] = c1; red[2][tid] = c2;
  __syncthreads();
  for (int s = 128; s > 0; s >>= 1) {
    if (tid < s)
      for (int q = 0; q < 3; ++q) red[q][tid] += red[q][tid + s];
    __syncthreads();
  }
  if (tid == 0)
    for (int q = 0; q < 3; ++q) bank[q * 64 + blockIdx.x] += red[q][0];
}

// --------------------------------------------------- triplet: anchor vector
__global__ __launch_bounds__(256) void anchor_kernel(const float* __restrict__ vaf_satt,
                                                     const int* __restrict__ label,
                                                     const int* __restrict__ seq_len,
                                                     float* __restrict__ anchor) {
  const int b = blockIdx.x;
  if (label[b] != 0) return;
  const int len = seq_len[b];
  const float* base = vaf_satt + (size_t)b * T_ * M_;
  const float inv = 1.f / (float)len;
  for (int j = threadIdx.x; j < M_; j += 256) {
    float s = 0.f;
    for (int t = 0; t < len; ++t) s += base[(size_t)t * M_ + j];
    atomicAdd(&anchor[j], s * inv);
  }
}

// ------------------------------------------------ triplet: pos/neg vectors
__global__ __launch_bounds__(256) void posneg_kernel(const float* __restrict__ vaf_satt,
                                                     const float* __restrict__ vaf_avf,
                                                     const int* __restrict__ label,
                                                     const int* __restrict__ seq_len,
                                                     float* __restrict__ pos,
                                                     float* __restrict__ neg) {
  const int b = blockIdx.x;
  if (label[b] != 1) return;
  const int len = seq_len[b];
  const int k = len / 16 + 1;
  __shared__ float sig[T_];
  __shared__ unsigned char inS[T_], inL[T_];
  const int t = threadIdx.x;       // blockDim == T_ == 256
  sig[t] = vaf_avf[b * T_ + t];
  __syncthreads();
  // stable rank: ascending by key, ties by index (matches jnp.argsort)
  const float kt  = (t < len) ?  sig[t] : __builtin_inff();
  const float ktL = (t < len) ? -sig[t] : __builtin_inff();
  int rs = 0, rl = 0;
  for (int j = 0; j < T_; ++j) {
    const float kj  = (j < len) ?  sig[j] : __builtin_inff();
    const float kjL = (j < len) ? -sig[j] : __builtin_inff();
    if (kj  < kt  || (kj  == kt  && j < t)) rs++;
    if (kjL < ktL || (kjL == ktL && j < t)) rl++;
  }
  inS[t] = (rs < k); inL[t] = (rl < k);
  __syncthreads();
  const float* base = vaf_satt + (size_t)b * T_ * M_;
  const float invk = 1.f / (float)k;
  for (int j = t; j < M_; j += 256) {
    float sp = 0.f, sn = 0.f;
    for (int tt = 0; tt < T_; ++tt) {
      const float v = base[(size_t)tt * M_ + j];
      if (inS[tt]) sp += v;
      if (inL[tt]) sn += v;
    }
    atomicAdd(&pos[j], sp * invk);
    atomicAdd(&neg[j], sn * invk);
  }
}

// -------------------------------------------------------------- finalize
__global__ __launch_bounds__(256) void finalize_kernel(const float* __restrict__ cosbank,
                                                       const float* __restrict__ cebank,
                                                       const float* __restrict__ anchor,
                                                       const float* __restrict__ pos,
                                                       const float* __restrict__ neg,
                                                       const float* __restrict__ va_out,
                                                       const float* __restrict__ vf_out,
                                                       const float* __restrict__ vaf_out,
                                                       const int* __restrict__ label,
                                                       const float* __restrict__ l1p,
                                                       const float* __restrict__ l2p,
                                                       const float* __restrict__ l3p,
                                                       const float* __restrict__ l4p,
                                                       float* __restrict__ out) {
  __shared__ float red[256];
  const int tid = threadIdx.x;
  auto bsum = [&](float v) -> float {
    red[tid] = v; __syncthreads();
    for (int s = 128; s > 0; s >>= 1) {
      if (tid < s) red[tid] += red[tid + s];
      __syncthreads();
    }
    float r = red[0]; __syncthreads();
    return r;
  };
  // cosine / CE bank sums
  float c0 = bsum(tid < 64 ? cosbank[0 * 64 + tid] : 0.f);
  float c1 = bsum(tid < 64 ? cosbank[1 * 64 + tid] : 0.f);
  float c2 = bsum(tid < 64 ? cosbank[2 * 64 + tid] : 0.f);
  float e0 = bsum(tid < 64 ? cebank[0 * 64 + tid] : 0.f);
  float e1 = bsum(tid < 64 ? cebank[1 * 64 + tid] : 0.f);
  float e2 = bsum(tid < 64 ? cebank[2 * 64 + tid] : 0.f);
  // triplet norms (nrm is scale-invariant, so unscaled sums are fine)
  float sA = 0, sP = 0, sN = 0;
  for (int j = tid; j < M_; j += 256) {
    const float a = anchor[j], p = pos[j], n = neg[j];
    sA += a * a; sP += p * p; sN += n * n;
  }
  sA = bsum(sA); sP = bsum(sP); sN = bsum(sN);
  const float ia = (sA > 0.f) ? 1.f / sqrtf(sA) : 0.f;
  const float ip = (sP > 0.f) ? 1.f / sqrtf(sP) : 0.f;
  const float in = (sN > 0.f) ? 1.f / sqrtf(sN) : 0.f;
  float d1 = 0, d2 = 0;
  for (int j = tid; j < M_; j += 256) {
    const float a = anchor[j] * ia, p = pos[j] * ip, n = neg[j] * in;
    const float x1 = a - p + 1e-6f;   // torch pairwise_distance eps
    const float x2 = a - n + 1e-6f;
    d1 += x1 * x1; d2 += x2 * x2;
  }
  d1 = bsum(d1); d2 = bsum(d2);
  // BCEs + label counts
  float bA = 0, bF = 0, bR = 0, nc = 0, ac = 0;
  if (tid < B_) {
    const float y = (float)label[tid];
    const float pa = va_out[tid], pf = vf_out[tid], pr = vaf_out[tid];
    bA = -(y * logf(pa) + (1.f - y) * logf(1.f - pa));
    bF = -(y * logf(pf) + (1.f - y) * logf(1.f - pf));
    bR = -(y * logf(pr) + (1.f - y) * logf(1.f - pr));
    nc = (label[tid] == 0) ? 1.f : 0.f;
    ac = (label[tid] == 1) ? 1.f : 0.f;
  }
  bA = bsum(bA) / (float)B_;
  bF = bsum(bF) / (float)B_;
  bR = bsum(bR) / (float)B_;
  nc = bsum(nc); ac = bsum(ac);
  if (tid == 0) {
    const float d_sum = (c0 + c1 + c2) / ((float)T_ * (float)B_);
    const float invN = 1.f / (float)(B_ * T_);
    const float ma = d_sum + e0 * invN + e1 * invN + e2 * invN;
    float trip = 0.f;
    if (nc > 0.f && ac > 0.f)
      trip = fmaxf(sqrtf(d1) - sqrtf(d2) + 5.f, 0.f);
    const float l1 = *l1p, l2 = *l2p, l3 = *l3p, l4 = *l4p;
    const float af_loss = bA + bF;
    out[0] = l1 * ma + l2 * af_loss + l3 * bR + l4 * trip;
    out[1] = ma;
    out[2] = af_loss;
    out[3] = bR;
    out[4] = trip;
  }
}

// ------------------------------------------------------------------ launch
extern "C" void kernel_launch(void* const* d_in, const int* in_sizes, int n_in,
                              void* d_out, int out_size, void* d_ws, size_t ws_size,
                              hipStream_t stream) {
  const float* v_satt   = (const float*)d_in[0];
  const float* va_satt  = (const float*)d_in[1];
  const float* vf_satt  = (const float*)d_in[2];
  const float* vaf_satt = (const float*)d_in[3];
  const float* v_avf    = (const float*)d_in[4];
  const float* va_avf   = (const float*)d_in[5];
  const float* vf_avf   = (const float*)d_in[6];
  const float* vaf_avf  = (const float*)d_in[7];
  const float* va_out   = (const float*)d_in[8];
  const float* vf_out   = (const float*)d_in[9];
  const float* vaf_out  = (const float*)d_in[10];
  const float* l1 = (const float*)d_in[11];
  const float* l2 = (const float*)d_in[12];
  const float* l3 = (const float*)d_in[13];
  const float* l4 = (const float*)d_in[14];
  const int* label   = (const int*)d_in[15];
  const int* seq_len = (const int*)d_in[16];
  float* out = (float*)d_out;

  // workspace layout
  float* simA = (float*)d_ws;            // 512*1024
  float* simF = simA + OM_ * M_;         // 512*1024
  float* ssV  = simF + OM_ * M_;         // 1024
  float* ssA  = ssV + M_;                // 512
  float* ssF  = ssA + OM_;               // 512
  float* anchor  = ssF + OM_;            // 1024
  float* pos     = anchor + M_;          // 1024
  float* neg     = pos + M_;             // 1024
  float* cosbank = neg + M_;             // 3*64
  float* cebank  = cosbank + 192;        // 3*64
  int*   extA = (int*)(cebank + 192);    // 1024
  int*   extF = extA + M_;               // 1024
  const int nzero = M_ + OM_ + OM_ + 3 * M_ + 192 + 192;  // ssV..cebank

  zero_kernel<<<8, 256, 0, stream>>>(ssV, nzero);

  colsumsq_kernel<<<dim3(M_ / 256, 16), 256, 0, stream>>>(v_satt, ssV, M_);
  colsumsq_kernel<<<dim3(OM_ / 256, 16), 256, 0, stream>>>(va_satt, ssA, OM_);
  colsumsq_kernel<<<dim3(OM_ / 256, 16), 256, 0, stream>>>(vf_satt, ssF, OM_);

  // 32x32 output per wave -> (512/32)*(1024/32) = 512 waves, 8 waves/block
  const int blocks = (OM_ / 32) * (M_ / 32) / 8;   // 64
  gemm_wmma_kernel<<<blocks, 256, 0, stream>>>(va_satt, v_satt, ssA, ssV, simA, OM_);
  gemm_wmma_kernel<<<blocks, 256, 0, stream>>>(vf_satt, v_satt, ssF, ssV, simF, OM_);

  match_kernel<<<2, 1024, 0, stream>>>(simA, simF, extA, extF);

  cos_loss_kernel<<<N_, 256, 0, stream>>>(v_satt, va_satt, vf_satt, extA, extF, cosbank);
  ce_kernel<<<N_ / 256, 256, 0, stream>>>(v_avf, va_avf, vf_avf, seq_len, cebank);
  anchor_kernel<<<B_, 256, 0, stream>>>(vaf_satt, label, seq_len, anchor);
  posneg_kernel<<<B_, 256, 0, stream>>>(vaf_satt, vaf_avf, label, seq_len, pos, neg);

  finalize_kernel<<<1, 256, 0, stream>>>(cosbank, cebank, anchor, pos, neg,
                                         va_out, vf_out, vaf_out, label,
                                         l1, l2, l3, l4, out);
}